// GNNBlock_61984968015928
// MI455X (gfx1250) — compile-verified
//
#include <hip/hip_runtime.h>

// GIN GNN block for MI455X (gfx1250, wave32).
// Pipeline:
//   zero -> degree histogram -> exclusive scan -> CSR fill
//   -> per-node gather aggregate (fused (1+eps)x + self-loop)    [no f32 atomics]
//   -> WMMA fp32 GEMM1 (+BN1 stats) -> BN1 fold
//   -> WMMA fp32 GEMM2 (BN1+ReLU fused on A, +BN2 stats) -> BN2 fold
//   -> BN2 apply in-place on d_out.

typedef float v2f __attribute__((ext_vector_type(2)));
typedef float v8f __attribute__((ext_vector_type(8)));

#define DIM   128
#define DIM2  256
#define BN_EPS 1e-5f
#define ROWTILES 10          // 16*10 = 160 rows per GEMM workgroup; 100000/160 = 625

// -------- stats scratch layout (floats) --------
// [0..255]    sum1      [256..511]  sumsq1
// [512..767]  scale1    [768..1023] shift1
// [1024..1151] sum2     [1152..1279] sumsq2
// [1280..1407] scale2   [1408..1535] shift2
#define STATS_FLOATS 2048

// ---------------------------------------------------------------------
// K0: zero stats + degree counts.
__global__ __launch_bounds__(256)
void zero_kernel(float* __restrict__ stats, int* __restrict__ counts, int n)
{
    int i = blockIdx.x * blockDim.x + threadIdx.x;
    if (i < STATS_FLOATS) stats[i] = 0.0f;
    if (i < n) counts[i] = 0;
}

// ---------------------------------------------------------------------
// K1: destination-degree histogram.
__global__ __launch_bounds__(256)
void hist_kernel(const int* __restrict__ ei, int* __restrict__ counts, int E)
{
    int e = blockIdx.x * blockDim.x + threadIdx.x;
    if (e >= E) return;
    atomicAdd(&counts[ei[E + e]], 1);
}

// ---------------------------------------------------------------------
// K2: single-block exclusive scan of counts -> offsets (n+1).
//     Also rewrites counts[i] = offsets[i] to serve as the fill cursor.
__global__ __launch_bounds__(1024)
void scan_kernel(int* __restrict__ counts, int* __restrict__ offsets, int n)
{
    __shared__ int buf[1024];
    __shared__ int carry;
    int tid = threadIdx.x;
    if (tid == 0) carry = 0;
    __syncthreads();
    for (int base = 0; base < n; base += 1024) {
        int i = base + tid;
        int v = (i < n) ? counts[i] : 0;
        buf[tid] = v;
        __syncthreads();
        #pragma unroll
        for (int off = 1; off < 1024; off <<= 1) {
            int t = (tid >= off) ? buf[tid - off] : 0;
            __syncthreads();
            buf[tid] += t;
            __syncthreads();
        }
        int excl = carry + buf[tid] - v;
        if (i < n) { offsets[i] = excl; counts[i] = excl; }
        int tilesum = buf[1023];
        __syncthreads();
        if (tid == 0) carry += tilesum;
        __syncthreads();
    }
    if (tid == 0) offsets[n] = carry;
}

// ---------------------------------------------------------------------
// K3: scatter edge ids into CSR slots (cursor = counts, preloaded with offsets).
__global__ __launch_bounds__(256)
void fill_kernel(const int* __restrict__ ei, int* __restrict__ cursor,
                 int* __restrict__ csr, int E)
{
    int e = blockIdx.x * blockDim.x + threadIdx.x;
    if (e >= E) return;
    int p = atomicAdd(&cursor[ei[E + e]], 1);
    csr[p] = e;
}

// ---------------------------------------------------------------------
// K4: one wave per node. Register-accumulate all incident messages + fused
//     (1+eps)*x + self-loop term; single store of h. No float atomics.
__global__ __launch_bounds__(256)
void aggregate_kernel(const float* __restrict__ x,
                      const int* __restrict__ ei,
                      const int* __restrict__ ea,
                      const float* __restrict__ ew,
                      const float* __restrict__ emb0,
                      const float* __restrict__ emb1,
                      const float* __restrict__ emb2,
                      const float* __restrict__ epsp,
                      const int* __restrict__ offsets,
                      const int* __restrict__ csr,
                      float* __restrict__ h,
                      int n, int E)
{
    int t = blockIdx.x * blockDim.x + threadIdx.x;
    int v = t >> 5;
    if (v >= n) return;
    int lane = t & 31;
    int d = lane * 4;
    float eps1 = 1.0f + epsp[0];

    float4 xv = *(const float4*)(x + (size_t)v * DIM + d);
    float4 s0 = *(const float4*)(emb0 + 5 * DIM + d);
    float4 s1 = *(const float4*)(emb1 + 7 * DIM + d);
    float4 s2 = *(const float4*)(emb2 + 0 * DIM + d);
    float4 acc;
    acc.x = eps1 * xv.x + fmaxf(xv.x + s0.x + s1.x + s2.x, 0.0f);
    acc.y = eps1 * xv.y + fmaxf(xv.y + s0.y + s1.y + s2.y, 0.0f);
    acc.z = eps1 * xv.z + fmaxf(xv.z + s0.z + s1.z + s2.z, 0.0f);
    acc.w = eps1 * xv.w + fmaxf(xv.w + s0.w + s1.w + s2.w, 0.0f);

    int beg = offsets[v];
    int end = offsets[v + 1];
    for (int j = beg; j < end; ++j) {
        int e = csr[j];
        int src = ei[e];
        int a0 = ea[3 * e + 0];
        int a1 = ea[3 * e + 1];
        int a2 = ea[3 * e + 2];
        float w = ew[e];
        float4 xs = *(const float4*)(x + (size_t)src * DIM + d);
        float4 e0 = *(const float4*)(emb0 + (size_t)a0 * DIM + d);
        float4 e1 = *(const float4*)(emb1 + (size_t)a1 * DIM + d);
        float4 e2 = *(const float4*)(emb2 + (size_t)a2 * DIM + d);
        acc.x += fmaxf(xs.x + e0.x + e1.x + e2.x, 0.0f) * w;
        acc.y += fmaxf(xs.y + e0.y + e1.y + e2.y, 0.0f) * w;
        acc.z += fmaxf(xs.z + e0.z + e1.z + e2.z, 0.0f) * w;
        acc.w += fmaxf(xs.w + e0.w + e1.w + e2.w, 0.0f) * w;
    }
    *(float4*)(h + (size_t)v * DIM + d) = acc;
}

// ---------------------------------------------------------------------
// K5: GEMM1  h1 = h[n,128] @ W1[128,256] + b1, plus BN1 column sum/sumsq.
// One workgroup = 160 rows x 256 cols (W1 staged once, reused 10x).
// W1 in LDS column-major, stride 132 (bank-conflict-free v2f reads).
#define LDS1 132
__global__ __launch_bounds__(256)
void gemm1_kernel(const float* __restrict__ h,
                  const float* __restrict__ W1,
                  const float* __restrict__ b1,
                  float* __restrict__ h1,
                  float* __restrict__ stats)
{
    __shared__ float lw[DIM2 * LDS1];   // 256 cols * 132 k-stride = 135168 B
    int tid = threadIdx.x;
    for (int i = tid; i < DIM * DIM2; i += 256) {
        int k = i >> 8;          // row of W1
        int c = i & 255;         // col of W1
        lw[c * LDS1 + k] = W1[i];
    }
    __syncthreads();

    int wave = tid >> 5;
    int lane = tid & 31;
    int m16  = lane & 15;
    int hh   = lane >> 4;                 // lane half: selects K pair
    int cb0 = wave * 32;
    int cb1 = cb0 + 16;
    const float* bp0 = &lw[(cb0 + m16) * LDS1];
    const float* bp1 = &lw[(cb1 + m16) * LDS1];
    float* sum1 = stats;
    float* sq1  = stats + 256;

    for (int rt = 0; rt < ROWTILES; ++rt) {
        int row_base = (blockIdx.x * ROWTILES + rt) * 16;
        // prefetch next row tile of A (gfx1250 global_prefetch_b8)
        if (rt + 1 < ROWTILES) {
            const char* pf = (const char*)(h + (size_t)(row_base + 16 + (lane >> 1)) * DIM)
                             + (lane & 1) * 256;
            __builtin_prefetch(pf, 0, 1);
        }
        const float* arow = h + (size_t)(row_base + m16) * DIM;

        v8f c0 = {};
        v8f c1 = {};
        for (int k0 = 0; k0 < DIM; k0 += 4) {
            int kk = k0 + 2 * hh;
            v2f a   = *(const v2f*)(arow + kk);
            v2f b0v = *(const v2f*)(bp0 + kk);
            v2f b1v = *(const v2f*)(bp1 + kk);
            c0 = __builtin_amdgcn_wmma_f32_16x16x4_f32(false, a, false, b0v,
                                                       (short)0, c0, false, false);
            c1 = __builtin_amdgcn_wmma_f32_16x16x4_f32(false, a, false, b1v,
                                                       (short)0, c1, false, false);
        }
        // tile 0
        {
            int col = cb0 + m16;
            float bias = b1[col];
            float s = 0.0f, q = 0.0f;
            #pragma unroll
            for (int r = 0; r < 8; ++r) {
                float vv = c0[r] + bias;
                h1[(size_t)(row_base + r + 8 * hh) * DIM2 + col] = vv;
                s += vv; q += vv * vv;
            }
            s += __shfl_xor(s, 16, 32);
            q += __shfl_xor(q, 16, 32);
            if (hh == 0) { unsafeAtomicAdd(&sum1[col], s); unsafeAtomicAdd(&sq1[col], q); }
        }
        // tile 1
        {
            int col = cb1 + m16;
            float bias = b1[col];
            float s = 0.0f, q = 0.0f;
            #pragma unroll
            for (int r = 0; r < 8; ++r) {
                float vv = c1[r] + bias;
                h1[(size_t)(row_base + r + 8 * hh) * DIM2 + col] = vv;
                s += vv; q += vv * vv;
            }
            s += __shfl_xor(s, 16, 32);
            q += __shfl_xor(q, 16, 32);
            if (hh == 0) { unsafeAtomicAdd(&sum1[col], s); unsafeAtomicAdd(&sq1[col], q); }
        }
    }
}

// ---------------------------------------------------------------------
// K6: fold BN1 into scale/shift.  (1 block, 256 threads)
__global__ void finalize1_kernel(const float* __restrict__ g,
                                 const float* __restrict__ b,
                                 float* __restrict__ stats, int n)
{
    int c = threadIdx.x;
    float inv_n = 1.0f / (float)n;
    float mean = stats[c] * inv_n;
    float var  = stats[256 + c] * inv_n - mean * mean;
    float sc   = g[c] * rsqrtf(var + BN_EPS);
    stats[512 + c] = sc;
    stats[768 + c] = b[c] - sc * mean;
}

// ---------------------------------------------------------------------
// K7: GEMM2  out = relu(bn1(h1)) @ W2[256,128] + b2, plus BN2 column stats.
// One workgroup = 160 rows x 128 cols; 8 waves, one 16x16 C tile each per row tile.
#define LDS2 260
__global__ __launch_bounds__(256)
void gemm2_kernel(const float* __restrict__ h1,
                  const float* __restrict__ W2,
                  const float* __restrict__ b2,
                  const float* __restrict__ stats,   // scale1/shift1 at 512/768
                  float* __restrict__ out,
                  float* __restrict__ stats_out)
{
    __shared__ float lw[DIM * LDS2];    // 128 cols * 260 k-stride = 133120 B
    __shared__ float lsc[DIM2];
    __shared__ float lsh[DIM2];
    int tid = threadIdx.x;
    for (int i = tid; i < DIM2 * DIM; i += 256) {
        int k = i >> 7;          // row of W2
        int c = i & 127;         // col of W2
        lw[c * LDS2 + k] = W2[i];
    }
    lsc[tid] = stats[512 + tid];
    lsh[tid] = stats[768 + tid];
    __syncthreads();

    int wave = tid >> 5;
    int lane = tid & 31;
    int m16  = lane & 15;
    int hh   = lane >> 4;
    int cb = wave * 16;
    const float* bp = &lw[(cb + m16) * LDS2];
    float* sum2 = stats_out + 1024;
    float* sq2  = stats_out + 1152;

    for (int rt = 0; rt < ROWTILES; ++rt) {
        int row_base = (blockIdx.x * ROWTILES + rt) * 16;
        if (rt + 1 < ROWTILES) {
            const char* pf = (const char*)(h1 + (size_t)(row_base + 16 + (lane >> 2)) * DIM2)
                             + (lane & 3) * 256;
            __builtin_prefetch(pf, 0, 1);
        }
        const float* arow = h1 + (size_t)(row_base + m16) * DIM2;

        v8f acc = {};
        for (int k0 = 0; k0 < DIM2; k0 += 4) {
            int kk = k0 + 2 * hh;
            v2f araw = *(const v2f*)(arow + kk);
            v2f a;
            a.x = fmaxf(lsc[kk]     * araw.x + lsh[kk],     0.0f);
            a.y = fmaxf(lsc[kk + 1] * araw.y + lsh[kk + 1], 0.0f);
            v2f bv = *(const v2f*)(bp + kk);
            acc = __builtin_amdgcn_wmma_f32_16x16x4_f32(false, a, false, bv,
                                                        (short)0, acc, false, false);
        }

        int col = cb + m16;
        float bias = b2[col];
        float s = 0.0f, q = 0.0f;
        #pragma unroll
        for (int r = 0; r < 8; ++r) {
            float vv = acc[r] + bias;
            out[(size_t)(row_base + r + 8 * hh) * DIM + col] = vv;
            s += vv; q += vv * vv;
        }
        s += __shfl_xor(s, 16, 32);
        q += __shfl_xor(q, 16, 32);
        if (hh == 0) { unsafeAtomicAdd(&sum2[col], s); unsafeAtomicAdd(&sq2[col], q); }
    }
}

// ---------------------------------------------------------------------
// K8: fold BN2 into scale/shift.  (1 block, 128 threads)
__global__ void finalize2_kernel(const float* __restrict__ g,
                                 const float* __restrict__ b,
                                 float* __restrict__ stats, int n)
{
    int c = threadIdx.x;
    float inv_n = 1.0f / (float)n;
    float mean = stats[1024 + c] * inv_n;
    float var  = stats[1152 + c] * inv_n - mean * mean;
    float sc   = g[c] * rsqrtf(var + BN_EPS);
    stats[1280 + c] = sc;
    stats[1408 + c] = b[c] - sc * mean;
}

// ---------------------------------------------------------------------
// K9: out = scale2*out + shift2, in place (out fully written by K7 first).
__global__ __launch_bounds__(256)
void bn2_apply_kernel(float* __restrict__ out,
                      const float* __restrict__ stats,
                      int n)
{
    int f = blockIdx.x * blockDim.x + threadIdx.x;   // float4 index
    int total = n * (DIM / 4);
    if (f >= total) return;
    int d = (f & (DIM / 4 - 1)) * 4;
    float4 v = ((float4*)out)[f];
    v.x = stats[1280 + d + 0] * v.x + stats[1408 + d + 0];
    v.y = stats[1280 + d + 1] * v.y + stats[1408 + d + 1];
    v.z = stats[1280 + d + 2] * v.z + stats[1408 + d + 2];
    v.w = stats[1280 + d + 3] * v.w + stats[1408 + d + 3];
    ((float4*)out)[f] = v;
}

// ---------------------------------------------------------------------
extern "C" void kernel_launch(void* const* d_in, const int* in_sizes, int n_in,
                              void* d_out, int out_size, void* d_ws, size_t ws_size,
                              hipStream_t stream)
{
    const float* x    = (const float*)d_in[0];
    const int*   ei   = (const int*)  d_in[1];
    const int*   ea   = (const int*)  d_in[2];
    const float* ew   = (const float*)d_in[3];
    const float* emb0 = (const float*)d_in[4];
    const float* emb1 = (const float*)d_in[5];
    const float* emb2 = (const float*)d_in[6];
    const float* eps  = (const float*)d_in[7];
    const float* W1   = (const float*)d_in[8];
    const float* b1   = (const float*)d_in[9];
    const float* bn1g = (const float*)d_in[10];
    const float* bn1b = (const float*)d_in[11];
    const float* W2   = (const float*)d_in[12];
    const float* b2   = (const float*)d_in[13];
    const float* bn2g = (const float*)d_in[14];
    const float* bn2b = (const float*)d_in[15];

    const int n = in_sizes[0] / DIM;   // 100000 (divisible by 160)
    const int E = in_sizes[3];         // 600000

    float* ws    = (float*)d_ws;
    float* h     = ws;                                // n*DIM
    float* h1    = h + (size_t)n * DIM;               // n*DIM2
    float* stats = h1 + (size_t)n * DIM2;             // STATS_FLOATS
    int*   counts  = (int*)(stats + STATS_FLOATS);    // n (histogram, then cursor)
    int*   offsets = counts + n;                      // n+1
    int*   csr     = offsets + n + 1;                 // E

    float* out = (float*)d_out;

    // K0: zero stats + counts
    {
        int threads = (n > STATS_FLOATS) ? n : STATS_FLOATS;
        zero_kernel<<<(threads + 255) / 256, 256, 0, stream>>>(stats, counts, n);
    }
    // K1: destination-degree histogram
    hist_kernel<<<(E + 255) / 256, 256, 0, stream>>>(ei, counts, E);
    // K2: exclusive scan -> offsets (+cursor in counts)
    scan_kernel<<<1, 1024, 0, stream>>>(counts, offsets, n);
    // K3: CSR fill
    fill_kernel<<<(E + 255) / 256, 256, 0, stream>>>(ei, counts, csr, E);
    // K4: per-node gather aggregate (fused init)
    {
        long long threads = (long long)n * 32;
        aggregate_kernel<<<(int)((threads + 255) / 256), 256, 0, stream>>>(
            x, ei, ea, ew, emb0, emb1, emb2, eps, offsets, csr, h, n, E);
    }
    // K5: GEMM1 + BN1 stats
    gemm1_kernel<<<n / (16 * ROWTILES), 256, 0, stream>>>(h, W1, b1, h1, stats);
    // K6: BN1 finalize
    finalize1_kernel<<<1, 256, 0, stream>>>(bn1g, bn1b, stats, n);
    // K7: GEMM2 (BN1+ReLU fused) + BN2 stats
    gemm2_kernel<<<n / (16 * ROWTILES), 256, 0, stream>>>(h1, W2, b2, stats, out, stats);
    // K8: BN2 finalize
    finalize2_kernel<<<1, 128, 0, stream>>>(bn2g, bn2b, stats, n);
    // K9: BN2 apply in place
    bn2_apply_kernel<<<(n * (DIM / 4) + 255) / 256, 256, 0, stream>>>(out, stats, n);
}